// CustomTransformer_60619168416497
// MI455X (gfx1250) — compile-verified
//
#include <hip/hip_runtime.h>

typedef __attribute__((ext_vector_type(2))) float v2f;
typedef __attribute__((ext_vector_type(8))) float v8f;

#define EPS 1e-7f

// ---------------------------------------------------------------------------
// Kernel A: global sum / sum-of-squares of X (f32 stream, f64 atomic accum)
// ---------------------------------------------------------------------------
__global__ void k_stats(const float* __restrict__ X, int n4, double* __restrict__ acc) {
    int idx    = blockIdx.x * blockDim.x + threadIdx.x;
    int stride = gridDim.x * blockDim.x;
    const float4* X4 = (const float4*)X;
    float s = 0.f, ss = 0.f;
    for (int i = idx; i < n4; i += stride) {
        float4 v = X4[i];
        s  += v.x + v.y + v.z + v.w;
        ss += v.x * v.x + v.y * v.y + v.z * v.z + v.w * v.w;
    }
    // wave32 reduction
    #pragma unroll
    for (int m = 16; m >= 1; m >>= 1) {
        s  += __shfl_xor(s,  m, 32);
        ss += __shfl_xor(ss, m, 32);
    }
    if ((threadIdx.x & 31) == 0) {
        atomicAdd(acc + 0, (double)s);
        atomicAdd(acc + 1, (double)ss);
    }
}

// ---------------------------------------------------------------------------
// Kernel B: fold all constant algebra.
//  pf[0]=mean pf[1]=invstd pf[2]=sA pf[3]=sB pf[4]=cls_score
//  pf[8..15]=w8raw (W1 @ W_k @ q_cls)   pf[16..47]=t32 (cls @ W_t)
// ---------------------------------------------------------------------------
__global__ void k_prep(const double* __restrict__ acc,
                       const float* __restrict__ W1, const float* __restrict__ cls,
                       const float* __restrict__ Wq, const float* __restrict__ Wk,
                       const float* __restrict__ Wt, float* __restrict__ pf,
                       long long n_elems) {
    int h = threadIdx.x;            // 0..31, one wave
    __shared__ float q[32], kkv[32];

    float qh = 0.f;                 // q_cls = cls @ W_q
    #pragma unroll
    for (int i = 0; i < 32; ++i) qh += cls[i] * Wq[i * 32 + h];
    q[h] = qh;
    __syncthreads();

    float kj = 0.f;                 // kk = W_k @ q_cls
    #pragma unroll
    for (int t = 0; t < 32; ++t) kj += Wk[h * 32 + t] * q[t];
    kkv[h] = kj;
    __syncthreads();

    float th = 0.f;                 // t32 = cls @ W_t
    #pragma unroll
    for (int i = 0; i < 32; ++i) th += cls[i] * Wt[i * 32 + h];
    pf[16 + h] = th;

    if (h == 0) {
        double N    = (double)n_elems;
        double mean = acc[0] / N;
        double var  = (acc[1] - acc[0] * acc[0] / N) / (N - 1.0);   // ddof=1
        float  inv  = (float)(1.0 / (sqrt(var) + (double)EPS));
        float  mf   = (float)mean;
        const float rs32 = 0.1767766952966369f;                     // 1/sqrt(32)
        float sumw = 0.f, csc = 0.f;
        for (int i = 0; i < 8; ++i) {
            float w = 0.f;
            for (int j = 0; j < 32; ++j) w += W1[i * 32 + j] * kkv[j];
            pf[8 + i] = w;
            sumw += w;
        }
        for (int j = 0; j < 32; ++j) csc += cls[j] * kkv[j];
        pf[0] = mf;
        pf[1] = inv;
        pf[2] = inv * rs32;               // score = dot*sA + sB
        pf[3] = -mf * sumw * inv * rs32;
        pf[4] = csc * rs32;               // raw CLS-row score (unnormalized row)
    }
}

// ---------------------------------------------------------------------------
// Kernel C: one block per batch. Stream X[b] (8 KB), softmax over 257,
// produce U[b][32] = (softmax-weighted xn) @ W1 + S_cls * cls
// ---------------------------------------------------------------------------
__global__ void k_main(const float* __restrict__ X, const float* __restrict__ pf,
                       const float* __restrict__ W1, const float* __restrict__ cls,
                       float* __restrict__ U) {
    const int b = blockIdx.x;
    const int k = threadIdx.x;          // 0..255
    const int wave = k >> 5, lane = k & 31;

    __shared__ float red[12];           // [0..7] wave partials, [8]=max, [9]=1/den, [10]=S_cls
    __shared__ float a8s[8][8];         // per-wave a8 partials
    __shared__ float fin[8];            // normalized a8

    const float mean = pf[0], invstd = pf[1], sA = pf[2], sB = pf[3], csc = pf[4];

    const float4* row = (const float4*)(X + ((size_t)b * 256 + k) * 8);
    float4 x0 = row[0], x1 = row[1];

    float d = x0.x * pf[8]  + x0.y * pf[9]  + x0.z * pf[10] + x0.w * pf[11]
            + x1.x * pf[12] + x1.y * pf[13] + x1.z * pf[14] + x1.w * pf[15];
    float score = d * sA + sB;

    // --- block max over 256 scores (+ cls) ---
    float m = score;
    #pragma unroll
    for (int msk = 16; msk >= 1; msk >>= 1) m = fmaxf(m, __shfl_xor(m, msk, 32));
    if (lane == 0) red[wave] = m;
    __syncthreads();
    if (k == 0) {
        float mm = csc;
        for (int i = 0; i < 8; ++i) mm = fmaxf(mm, red[i]);
        red[8] = mm;
    }
    __syncthreads();
    const float M = red[8];

    // --- block sum ---
    float p = __expf(score - M);
    float s = p;
    #pragma unroll
    for (int msk = 16; msk >= 1; msk >>= 1) s += __shfl_xor(s, msk, 32);
    if (lane == 0) red[wave] = s;
    __syncthreads();
    if (k == 0) {
        float pcls = __expf(csc - M);
        float den  = pcls;
        for (int i = 0; i < 8; ++i) den += red[i];
        red[9]  = 1.0f / den;
        red[10] = pcls / den;           // S_cls
    }
    __syncthreads();

    const float Sk = p * red[9];

    // --- a8 = sum_k Sk * X[b,k,:] (8 components) ---
    float av[8] = { Sk * x0.x, Sk * x0.y, Sk * x0.z, Sk * x0.w,
                    Sk * x1.x, Sk * x1.y, Sk * x1.z, Sk * x1.w };
    #pragma unroll
    for (int i = 0; i < 8; ++i) {
        float t = av[i];
        #pragma unroll
        for (int msk = 16; msk >= 1; msk >>= 1) t += __shfl_xor(t, msk, 32);
        if (lane == 0) a8s[wave][i] = t;
    }
    __syncthreads();
    if (k < 8) {
        float t = 0.f;
        for (int w = 0; w < 8; ++w) t += a8s[w][k];
        float psum = 1.0f - red[10];                      // sum of data-row weights
        fin[k] = (t - psum * mean) * invstd;              // normalized a8
    }
    __syncthreads();
    if (k < 32) {
        const float Scls = red[10];
        float u = Scls * cls[k];
        #pragma unroll
        for (int i = 0; i < 8; ++i) u += fin[i] * W1[i * 32 + k];
        U[(size_t)b * 32 + k] = u;
    }
}

// ---------------------------------------------------------------------------
// Kernel D: 16 batches per wave. H2 = U_tile(16x32) @ W_v(32x32) + t32 via
// V_WMMA_F32_16X16X4_F32 (two 16x16 N-tiles, 8 K-steps), then logits + NLL.
// ---------------------------------------------------------------------------
__global__ void k_final(const float* __restrict__ U, const float* __restrict__ pf,
                        const float* __restrict__ Wv, const float* __restrict__ W2,
                        const long long* __restrict__ y, float* __restrict__ out) {
    const int lane = threadIdx.x;          // single wave of 32
    const int b0   = blockIdx.x * 16;

    v8f acc0 = {}, acc1 = {};
    const int row  = b0 + (lane & 15);     // A: lanes 0-15 -> K0,K1 ; 16-31 -> K2,K3
    const int ksel = (lane >> 4) * 2;
    const int n    = lane & 15;            // B: lanes 0-15 -> col n ; 16-31 -> col n (hi K)

    #pragma unroll
    for (int s = 0; s < 8; ++s) {
        const int kb = 4 * s + ksel;
        v2f a;  a[0]  = U[(size_t)row * 32 + kb];   a[1]  = U[(size_t)row * 32 + kb + 1];
        v2f bl; bl[0] = Wv[kb * 32 + n];            bl[1] = Wv[(kb + 1) * 32 + n];
        v2f bh; bh[0] = Wv[kb * 32 + 16 + n];       bh[1] = Wv[(kb + 1) * 32 + 16 + n];
        acc0 = __builtin_amdgcn_wmma_f32_16x16x4_f32(false, a, false, bl,
                                                     (short)0, acc0, false, false);
        acc1 = __builtin_amdgcn_wmma_f32_16x16x4_f32(false, a, false, bh,
                                                     (short)0, acc1, false, false);
    }

    // D layout: VGPR r, lanes 0-15 -> (row r, col lane); lanes 16-31 -> (row 8+r, col lane-16)
    __shared__ float h2[16][33];
    const float* t32 = pf + 16;
    const int colA  = lane & 15;
    const int rbase = (lane >> 4) * 8;
    #pragma unroll
    for (int r = 0; r < 8; ++r) {
        h2[rbase + r][colA]      = acc0[r] + t32[colA];
        h2[rbase + r][colA + 16] = acc1[r] + t32[colA + 16];
    }
    __syncthreads();

    if (lane < 16) {
        const int b = b0 + lane;
        float z0 = 0.f, z1 = 0.f;
        #pragma unroll
        for (int c = 0; c < 32; ++c) {
            float h = h2[lane][c];
            z0 += h * W2[c * 2 + 0];
            z1 += h * W2[c * 2 + 1];
        }
        float mz  = fmaxf(z0, z1);
        float lse = mz + __logf(__expf(z0 - mz) + __expf(z1 - mz));
        float zt  = (y[b] != 0) ? z1 : z0;
        atomicAdd(out, (lse - zt) * (1.0f / 2048.0f));
    }
}

// ---------------------------------------------------------------------------
extern "C" void kernel_launch(void* const* d_in, const int* in_sizes, int n_in,
                              void* d_out, int out_size, void* d_ws, size_t ws_size,
                              hipStream_t stream) {
    const float*     X   = (const float*)d_in[0];
    const long long* y   = (const long long*)d_in[1];
    const float*     W1  = (const float*)d_in[2];
    const float*     cls = (const float*)d_in[3];
    const float*     Wq  = (const float*)d_in[4];
    const float*     Wk  = (const float*)d_in[5];
    const float*     Wv  = (const float*)d_in[6];
    const float*     Wt  = (const float*)d_in[7];
    const float*     W2  = (const float*)d_in[8];
    float*           out = (float*)d_out;

    char*   ws  = (char*)d_ws;
    double* acc = (double*)ws;              // 2 doubles
    float*  pf  = (float*)(ws + 64);        // 48 floats of folded constants
    float*  U   = (float*)(ws + 1024);      // [B x 32]

    const int nX = in_sizes[0];             // 2048*256*8
    const int B  = nX / (256 * 8);          // 2048

    hipMemsetAsync(acc, 0, 2 * sizeof(double), stream);
    hipMemsetAsync(out, 0, sizeof(float), stream);

    k_stats<<<1024, 256, 0, stream>>>(X, nX / 4, acc);
    k_prep <<<1,    32,  0, stream>>>(acc, W1, cls, Wq, Wk, Wt, pf, (long long)nX);
    k_main <<<B,    256, 0, stream>>>(X, pf, W1, cls, U);
    k_final<<<B/16, 32,  0, stream>>>(U, pf, Wv, W2, y, out);
}